// CloudResourceGNNv1_45964740002546
// MI455X (gfx1250) — compile-verified
//
#include <hip/hip_runtime.h>

typedef __attribute__((ext_vector_type(2))) float v2f;
typedef __attribute__((ext_vector_type(8))) float v8f;

#define NEG_SLOPE 0.2f
#define LN_EPS 1e-5f

__device__ __forceinline__ float elu1(float x) { return x > 0.f ? x : (__expf(x) - 1.f); }
__device__ __forceinline__ float lrelu(float x) { return x > 0.f ? x : NEG_SLOPE * x; }

// float atomic max via integer ordering trick (maps to global_atomic_max/min)
__device__ __forceinline__ void atomicMaxF(float* a, float v) {
  if (v >= 0.f) atomicMax((int*)a, __float_as_int(v));
  else          atomicMin((unsigned int*)a, __float_as_uint(v));
}

__global__ void fill_f32(float* __restrict__ p, float v, int n) {
  int i = blockIdx.x * blockDim.x + threadIdx.x;
  if (i < n) p[i] = v;
}

// ---------------------------------------------------------------------------
// Generic WMMA f32 GEMM: C[M,NC] = A[M,K] * B[K,NC]
// One wave computes one 16x16 tile via V_WMMA_F32_16X16X4_F32.
// Block = NC/16 waves sharing one 16-row A panel staged in LDS.
// M must be a multiple of 16 (50000 and 800000 both are) -> EXEC all ones.
// ---------------------------------------------------------------------------
template <int K, int NC>
__global__ void gemm_wmma_f32(const float* __restrict__ A, const float* __restrict__ B,
                              float* __restrict__ C) {
  constexpr int NW = NC / 16;
  __shared__ float Al[16][K + 4];
  const int mbase = blockIdx.x * 16;
  const int tid = threadIdx.x;
  for (int i = tid; i < 16 * K; i += 32 * NW) {
    int r = i / K, c = i - r * K;
    Al[r][c] = A[(size_t)(mbase + r) * K + c];
  }
  __syncthreads();
  const int wave = tid >> 5, lane = tid & 31;
  const int half = lane >> 4, m = lane & 15;
  const int ncol = wave * 16 + m;
  v8f acc = {};
#pragma unroll
  for (int k0 = 0; k0 < K; k0 += 4) {
    // A frag: lanes 0-15 -> M=m, K=k0..k0+1 ; lanes 16-31 -> K=k0+2..k0+3
    v2f a = *(const v2f*)&Al[m][k0 + 2 * half];
    v2f b;
    b.x = B[(size_t)(k0 + 2 * half) * NC + ncol];
    b.y = B[(size_t)(k0 + 2 * half + 1) * NC + ncol];
    acc = __builtin_amdgcn_wmma_f32_16x16x4_f32(false, a, false, b, (short)0, acc,
                                                false, false);
  }
#pragma unroll
  for (int i = 0; i < 8; ++i)
    C[(size_t)(mbase + half * 8 + i) * NC + ncol] = acc[i];
}

// Resource GEMM: rows = N*R, K=64, 64 output cols; fused +bias, ELU,
// written to the RIGHT half (cols 64..127) of the [N*R,128] output.
__global__ void gemm_res_wmma(const float* __restrict__ RF, const float* __restrict__ W,
                              const float* __restrict__ bias, float* __restrict__ out) {
  constexpr int K = 64;
  __shared__ float Al[16][K + 4];
  const int mbase = blockIdx.x * 16;
  const int tid = threadIdx.x;  // 128 threads = 4 waves
  for (int i = tid; i < 16 * K; i += 128) {
    int r = i >> 6, c = i & 63;
    Al[r][c] = RF[(size_t)(mbase + r) * K + c];
  }
  __syncthreads();
  const int wave = tid >> 5, lane = tid & 31;
  const int half = lane >> 4, m = lane & 15;
  const int ncol = wave * 16 + m;
  v8f acc = {};
#pragma unroll
  for (int k0 = 0; k0 < K; k0 += 4) {
    v2f a = *(const v2f*)&Al[m][k0 + 2 * half];
    v2f b;
    b.x = W[(k0 + 2 * half) * 64 + ncol];
    b.y = W[(k0 + 2 * half + 1) * 64 + ncol];
    acc = __builtin_amdgcn_wmma_f32_16x16x4_f32(false, a, false, b, (short)0, acc,
                                                false, false);
  }
  const float bb = bias[ncol];
#pragma unroll
  for (int i = 0; i < 8; ++i)
    out[(size_t)(mbase + half * 8 + i) * 128 + 64 + ncol] = elu1(acc[i] + bb);
}

// Per-node attention coefficients: a_src[n,h] = dot(h[n,h,:], att_src[h,:]) (wave/node)
template <int H>
__global__ void attn_coeff(const float* __restrict__ h, const float* __restrict__ as_w,
                           const float* __restrict__ ad_w, float* __restrict__ as,
                           float* __restrict__ ad, int N) {
  const int n = (blockIdx.x * blockDim.x + threadIdx.x) >> 5;
  const int lane = threadIdx.x & 31;
  if (n >= N) return;
  if (H == 2) {
    const int head = lane >> 4, l = lane & 15;
    float ss = 0.f, dd = 0.f;
#pragma unroll
    for (int i = 0; i < 4; ++i) {
      int c = head * 64 + l * 4 + i;
      float v = h[(size_t)n * 128 + c];
      ss += v * as_w[c];
      dd += v * ad_w[c];
    }
#pragma unroll
    for (int off = 8; off >= 1; off >>= 1) {
      ss += __shfl_xor(ss, off, 16);
      dd += __shfl_xor(dd, off, 16);
    }
    if (l == 0) { as[n * 2 + head] = ss; ad[n * 2 + head] = dd; }
  } else {
    float ss = 0.f, dd = 0.f;
#pragma unroll
    for (int i = 0; i < 2; ++i) {
      int c = lane * 2 + i;
      float v = h[(size_t)n * 64 + c];
      ss += v * as_w[c];
      dd += v * ad_w[c];
    }
#pragma unroll
    for (int off = 16; off >= 1; off >>= 1) {
      ss += __shfl_xor(ss, off, 32);
      dd += __shfl_xor(dd, off, 32);
    }
    if (lane == 0) { as[n] = ss; ad[n] = dd; }
  }
}

__device__ __forceinline__ void edge_sd(int e, int E, const int* src, const int* dst,
                                        int& s, int& d) {
  if (e < E) { s = src[e]; d = dst[e]; }
  else       { s = e - E;  d = e - E; }   // self loop
}

template <int H>
__global__ void edge_max(const int* __restrict__ src, const int* __restrict__ dst,
                         const float* __restrict__ as, const float* __restrict__ ad,
                         float* __restrict__ emax, int E, int Etot) {
  int e = blockIdx.x * blockDim.x + threadIdx.x;
  if (e >= Etot) return;
  int s, d; edge_sd(e, E, src, dst, s, d);
#pragma unroll
  for (int h = 0; h < H; ++h) {
    float v = lrelu(as[(size_t)s * H + h] + ad[(size_t)d * H + h]);
    atomicMaxF(&emax[(size_t)d * H + h], v);
  }
}

template <int H>
__global__ void edge_exsum(const int* __restrict__ src, const int* __restrict__ dst,
                           const float* __restrict__ as, const float* __restrict__ ad,
                           const float* __restrict__ emax, float* __restrict__ exbuf,
                           float* __restrict__ denom, int E, int Etot) {
  int e = blockIdx.x * blockDim.x + threadIdx.x;
  if (e >= Etot) return;
  int s, d; edge_sd(e, E, src, dst, s, d);
#pragma unroll
  for (int h = 0; h < H; ++h) {
    float v = lrelu(as[(size_t)s * H + h] + ad[(size_t)d * H + h]);
    float ex = __expf(v - emax[(size_t)d * H + h]);
    exbuf[(size_t)e * H + h] = ex;
    atomicAdd(&denom[(size_t)d * H + h], ex);
  }
}

// wave per edge; H*C channels, channel c = i*32 + lane (coalesced)
template <int H, int C>
__global__ void edge_agg(const int* __restrict__ src, const int* __restrict__ dst,
                         const float* __restrict__ hsrc, const float* __restrict__ exbuf,
                         const float* __restrict__ denom, float* __restrict__ agg,
                         int E, int Etot) {
  int e = (blockIdx.x * blockDim.x + threadIdx.x) >> 5;
  int lane = threadIdx.x & 31;
  if (e >= Etot) return;
  int s, d; edge_sd(e, E, src, dst, s, d);
  float alpha[H];
#pragma unroll
  for (int h = 0; h < H; ++h)
    alpha[h] = exbuf[(size_t)e * H + h] / denom[(size_t)d * H + h];
  constexpr int PER = (H * C) / 32;
#pragma unroll
  for (int i = 0; i < PER; ++i) {
    int c = i * 32 + lane;
    int h = c / C;
    atomicAdd(&agg[(size_t)d * (H * C) + c],
              hsrc[(size_t)s * (H * C) + c] * alpha[h]);
  }
}

// h1out = elu(agg + b1), in place, 128 channels
__global__ void bias_elu128(float* __restrict__ a, const float* __restrict__ b, int total) {
  int i = blockIdx.x * blockDim.x + threadIdx.x;
  if (i >= total) return;
  a[i] = elu1(a[i] + b[i & 127]);
}

// LayerNorm over 64 channels: wave per node, 2 ch per lane
__global__ void ln_finish(const float* __restrict__ agg, const float* __restrict__ b2,
                          const float* __restrict__ w, const float* __restrict__ bb,
                          float* __restrict__ outh, int N) {
  int n = (blockIdx.x * blockDim.x + threadIdx.x) >> 5;
  int lane = threadIdx.x & 31;
  if (n >= N) return;
  float v0 = agg[(size_t)n * 64 + lane] + b2[lane];
  float v1 = agg[(size_t)n * 64 + 32 + lane] + b2[32 + lane];
  float s = v0 + v1;
#pragma unroll
  for (int off = 16; off >= 1; off >>= 1) s += __shfl_xor(s, off, 32);
  float mu = s * (1.f / 64.f);
  float d0 = v0 - mu, d1 = v1 - mu;
  float q = d0 * d0 + d1 * d1;
#pragma unroll
  for (int off = 16; off >= 1; off >>= 1) q += __shfl_xor(q, off, 32);
  float rs = rsqrtf(q * (1.f / 64.f) + LN_EPS);
  outh[(size_t)n * 64 + lane]      = d0 * rs * w[lane] + bb[lane];
  outh[(size_t)n * 64 + 32 + lane] = d1 * rs * w[32 + lane] + bb[32 + lane];
}

// out[m,0:64] = hfinal[m/16, :]  (left half of concat), coalesced
__global__ void bcast_left(const float* __restrict__ hfin, float* __restrict__ out,
                           int total /* (N*R)*64 */) {
  int i = blockIdx.x * blockDim.x + threadIdx.x;
  if (i >= total) return;
  int m = i >> 6, c = i & 63;
  out[(size_t)m * 128 + c] = hfin[(size_t)(m >> 4) * 64 + c];
}

extern "C" void kernel_launch(void* const* d_in, const int* in_sizes, int n_in,
                              void* d_out, int out_size, void* d_ws, size_t ws_size,
                              hipStream_t stream) {
  const float* x    = (const float*)d_in[0];
  const int*   ei   = (const int*)d_in[1];
  const float* rf   = (const float*)d_in[2];
  const float* W1   = (const float*)d_in[3];
  const float* asw1 = (const float*)d_in[4];
  const float* adw1 = (const float*)d_in[5];
  const float* b1   = (const float*)d_in[6];
  const float* W2   = (const float*)d_in[7];
  const float* asw2 = (const float*)d_in[8];
  const float* adw2 = (const float*)d_in[9];
  const float* b2   = (const float*)d_in[10];
  const float* lnw  = (const float*)d_in[11];
  const float* lnb  = (const float*)d_in[12];
  const float* resW = (const float*)d_in[13];
  const float* resb = (const float*)d_in[14];
  float* out = (float*)d_out;

  const int N = in_sizes[0] / 128;    // 50000
  const int E = in_sizes[1] / 2;      // 800000
  const int Etot = E + N;             // + self loops
  const int* src = ei;
  const int* dst = ei + E;

  // workspace carve-up (floats)
  float* ws = (float*)d_ws;
  float* h1   = ws; ws += (size_t)N * 128;
  float* agg1 = ws; ws += (size_t)N * 128;   // becomes h1out in-place
  float* as1  = ws; ws += (size_t)N * 2;
  float* ad1  = ws; ws += (size_t)N * 2;
  float* em1  = ws; ws += (size_t)N * 2;
  float* dn1  = ws; ws += (size_t)N * 2;
  float* ex1  = ws; ws += (size_t)Etot * 2;
  float* h2   = ws; ws += (size_t)N * 64;
  float* agg2 = ws; ws += (size_t)N * 64;
  float* as2  = ws; ws += (size_t)N;
  float* ad2  = ws; ws += (size_t)N;
  float* em2  = ws; ws += (size_t)N;
  float* dn2  = ws; ws += (size_t)N;
  float* ex2  = ws; ws += (size_t)Etot;
  float* hf   = ws; ws += (size_t)N * 64;

  hipMemsetAsync(agg1, 0, (size_t)N * 128 * sizeof(float), stream);
  hipMemsetAsync(agg2, 0, (size_t)N * 64 * sizeof(float), stream);
  hipMemsetAsync(dn1, 0, (size_t)N * 2 * sizeof(float), stream);
  hipMemsetAsync(dn2, 0, (size_t)N * sizeof(float), stream);
  fill_f32<<<(N * 2 + 255) / 256, 256, 0, stream>>>(em1, -3.0e38f, N * 2);
  fill_f32<<<(N + 255) / 256, 256, 0, stream>>>(em2, -3.0e38f, N);

  // ---- GAT layer 1 (heads=2, C=64, concat) ----
  gemm_wmma_f32<128, 128><<<N / 16, 256, 0, stream>>>(x, W1, h1);
  attn_coeff<2><<<(N * 32 + 255) / 256, 256, 0, stream>>>(h1, asw1, adw1, as1, ad1, N);
  edge_max<2><<<(Etot + 255) / 256, 256, 0, stream>>>(src, dst, as1, ad1, em1, E, Etot);
  edge_exsum<2><<<(Etot + 255) / 256, 256, 0, stream>>>(src, dst, as1, ad1, em1, ex1, dn1, E, Etot);
  edge_agg<2, 64><<<(Etot * 32 + 255) / 256, 256, 0, stream>>>(src, dst, h1, ex1, dn1, agg1, E, Etot);
  bias_elu128<<<(N * 128 + 255) / 256, 256, 0, stream>>>(agg1, b1, N * 128);

  // ---- GAT layer 2 (heads=1, C=64, mean==identity) ----
  gemm_wmma_f32<128, 64><<<N / 16, 128, 0, stream>>>(agg1, W2, h2);
  attn_coeff<1><<<(N * 32 + 255) / 256, 256, 0, stream>>>(h2, asw2, adw2, as2, ad2, N);
  edge_max<1><<<(Etot + 255) / 256, 256, 0, stream>>>(src, dst, as2, ad2, em2, E, Etot);
  edge_exsum<1><<<(Etot + 255) / 256, 256, 0, stream>>>(src, dst, as2, ad2, em2, ex2, dn2, E, Etot);
  edge_agg<1, 64><<<(Etot * 32 + 255) / 256, 256, 0, stream>>>(src, dst, h2, ex2, dn2, agg2, E, Etot);
  ln_finish<<<(N * 32 + 255) / 256, 256, 0, stream>>>(agg2, b2, lnw, lnb, hf, N);

  // ---- Resource MLP + concat ----
  gemm_res_wmma<<<N, 128, 0, stream>>>(rf, resW, resb, out);   // rows = N*16, /16 = N blocks
  bcast_left<<<(N * 16 * 64 + 255) / 256, 256, 0, stream>>>(hf, out, N * 16 * 64);
}